// GCN_22668837388509
// MI455X (gfx1250) — compile-verified
//
#include <hip/hip_runtime.h>
#include <cstdint>

// ---------------------------------------------------------------------------
// 2-layer GCN forward, fused pipeline for gfx1250 (MI455X), wave32.
//   deg -> dinv -> GEMM1(x@W1, f32 WMMA, async-LDS double-buffered x stream,
//   +self-loop init) -> edge agg1 -> GEMM2(relu(agg1+b1)@W2, f32 WMMA,
//   +self-loop init into d_out) -> edge agg2 -> (+b2, log_softmax) on d_out.
// ---------------------------------------------------------------------------

typedef __attribute__((ext_vector_type(2))) float v2f;
typedef __attribute__((ext_vector_type(8))) float v8f;

#define IN_C   512
#define HID_C  16
#define OUT_C  40
#define OUT_P  48   // OUT_C padded to 3 WMMA column tiles

#define KC     16   // K-chunk per async stage
#define APAD   20   // padded row stride (floats) -> conflict-free mod 64
#define ABUF_B (16 * APAD * 4)   // bytes per A buffer (1280)

__device__ __forceinline__ void atomAddF(float* p, float v) {
    unsafeAtomicAdd(p, v);   // global_atomic_add_f32 (L2-side atomics)
}

// gfx1250 async global->LDS copy, 16B per lane, tracked on ASYNCcnt.
__device__ __forceinline__ void async_ld_b128(uint32_t lds_off, const float* gaddr) {
    asm volatile("global_load_async_to_lds_b128 %0, %1, off"
                 :: "v"(lds_off), "v"(gaddr) : "memory");
}

// ---- degree / normalization ------------------------------------------------
__global__ __launch_bounds__(256) void k_deg_init(float* __restrict__ deg, int n) {
    int i = blockIdx.x * blockDim.x + threadIdx.x;
    if (i < n) deg[i] = 1.0f;                       // self-loop counts once
}

__global__ __launch_bounds__(256) void k_deg_count(const long long* __restrict__ ei,
                                                   float* __restrict__ deg, int E) {
    int i = blockIdx.x * blockDim.x + threadIdx.x;
    if (i < E) atomAddF(&deg[(int)ei[(size_t)E + i]], 1.0f);   // col = targets
}

__global__ __launch_bounds__(256) void k_rsqrt(float* __restrict__ deg, int n) {
    int i = blockIdx.x * blockDim.x + threadIdx.x;
    if (i < n) deg[i] = rsqrtf(deg[i]);             // deg >= 1 always
}

// ---- GEMM1: h1[N,16] = x[N,512] @ W1[512,16]; agg1 init = dinv^2 * h1 ------
// 8 waves/block, one 16-row tile per wave. x is streamed through LDS with
// double-buffered GLOBAL_LOAD_ASYNC_TO_LDS_B128 (ASYNCcnt), no block barriers
// in the K loop (each wave owns its LDS slab).
__global__ __launch_bounds__(256)
void k_gemm1(const float* __restrict__ x, const float* __restrict__ W1,
             const float* __restrict__ dinv, float* __restrict__ h1,
             float* __restrict__ agg1, int nrows)
{
    __shared__ float sW[IN_C * HID_C];         // 32 KB: W1[k][n]
    __shared__ float sA[8][2][16 * APAD];      // 20 KB: per-wave double buffer

    const int tid  = threadIdx.x;
    const int wave = tid >> 5;
    const int lane = tid & 31;
    const int r16  = lane & 15;
    const int kof  = (lane >> 4) << 1;         // 0 (lanes 0-15) / 2 (lanes 16-31)
    const int m0   = blockIdx.x * 128 + wave * 16;
    const bool active = (m0 < nrows);          // wave-uniform
    const int mbase = active ? m0 : (nrows - 16);   // clamp: loads always valid

    for (int i = tid; i < IN_C * HID_C; i += 256) sW[i] = W1[i];
    __syncthreads();

    // per-lane async copy geometry: 2 instrs/chunk, lane i covers (row, 4 cols)
    const int ar0 = (lane) >> 2;               // instr 0: rows 0..7
    const int ac0 = ((lane) & 3) << 2;
    const int ar1 = (32 + lane) >> 2;          // instr 1: rows 8..15
    const int ac1 = ac0;
    const uint32_t aBase = (uint32_t)(uintptr_t)(&sA[wave][0][0]);
    const float* xw = x + (size_t)mbase * IN_C;

    // prologue: stage chunk 0 into buffer 0
    async_ld_b128(aBase + (uint32_t)(ar0 * APAD + ac0) * 4, xw + (size_t)ar0 * IN_C + ac0);
    async_ld_b128(aBase + (uint32_t)(ar1 * APAD + ac1) * 4, xw + (size_t)ar1 * IN_C + ac1);

    v8f acc = {};
    for (int c = 0; c < IN_C / KC; ++c) {
        const int cur = c & 1;
        if (c + 1 < IN_C / KC) {
            const int nxt = cur ^ 1;
            const int k0n = (c + 1) * KC;
            // prior DS reads of 'nxt' buffer must retire before async overwrite
            asm volatile("s_wait_dscnt 0x0" ::: "memory");
            async_ld_b128(aBase + (uint32_t)nxt * ABUF_B + (uint32_t)(ar0 * APAD + ac0) * 4,
                          xw + (size_t)ar0 * IN_C + k0n + ac0);
            async_ld_b128(aBase + (uint32_t)nxt * ABUF_B + (uint32_t)(ar1 * APAD + ac1) * 4,
                          xw + (size_t)ar1 * IN_C + k0n + ac1);
            // chunk c complete when only the 2 just-issued ops remain (in-order)
            asm volatile("s_wait_asynccnt 0x2" ::: "memory");
        } else {
            asm volatile("s_wait_asynccnt 0x0" ::: "memory");
        }
        const int k0 = c * KC;
        #pragma unroll
        for (int kk = 0; kk < KC; kk += 4) {
            v2f a, b;
            a.x = sA[wave][cur][r16 * APAD + kk + kof];
            a.y = sA[wave][cur][r16 * APAD + kk + kof + 1];
            b.x = sW[(k0 + kk + kof) * HID_C + r16];
            b.y = sW[(k0 + kk + kof + 1) * HID_C + r16];
            acc = __builtin_amdgcn_wmma_f32_16x16x4_f32(false, a, false, b,
                                                        (short)0, acc, false, false);
        }
    }

    if (active) {
        const int row_off = (lane >> 4) << 3;       // 0 or 8
        #pragma unroll
        for (int g = 0; g < 8; ++g) {
            int   row = m0 + g + row_off;
            float v   = acc[g];
            float di  = dinv[row];
            h1[(size_t)row * HID_C + r16]   = v;
            agg1[(size_t)row * HID_C + r16] = di * di * v;   // self-loop term
        }
    }
}

// ---- edge aggregation, layer 1: agg1[col] += dinv[row]*dinv[col]*h1[row] ---
__global__ __launch_bounds__(256)
void k_agg1(const long long* __restrict__ ei, const float* __restrict__ dinv,
            const float* __restrict__ h1, float* __restrict__ agg1, int E)
{
    const int total  = E * HID_C;
    const int stride = gridDim.x * blockDim.x;
    for (int idx = blockIdx.x * blockDim.x + threadIdx.x; idx < total; idx += stride) {
        int e = idx >> 4, c = idx & 15;
        int r = (int)ei[e];
        int t = (int)ei[(size_t)E + e];
        float nrm = dinv[r] * dinv[t];
        atomAddF(&agg1[(size_t)t * HID_C + c], nrm * h1[(size_t)r * HID_C + c]);
    }
}

// ---- GEMM2: h2[N,40] = relu(agg1+b1) @ W2[16,40]; d_out init = dinv^2*h2 ---
__global__ __launch_bounds__(256)
void k_gemm2(const float* __restrict__ agg1, const float* __restrict__ b1,
             const float* __restrict__ W2, const float* __restrict__ dinv,
             float* __restrict__ h2, float* __restrict__ out, int nrows)
{
    __shared__ float sW[HID_C * OUT_P];     // 3 KB, zero-padded cols 40..47

    const int tid  = threadIdx.x;
    const int wave = tid >> 5;
    const int lane = tid & 31;
    const int r16  = lane & 15;
    const int kof  = (lane >> 4) << 1;
    const int m0   = blockIdx.x * 128 + wave * 16;

    for (int i = tid; i < HID_C * OUT_P; i += 256) {
        int r = i / OUT_P, c = i - r * OUT_P;
        sW[i] = (c < OUT_C) ? W2[r * OUT_C + c] : 0.0f;
    }
    __syncthreads();
    if (m0 >= nrows) return;                // uniform per wave; no syncs follow

    // A fragments for all 4 K-steps: relu(agg1 + b1), reused across 3 N-tiles
    v2f a[4];
    #pragma unroll
    for (int s = 0; s < 4; ++s) {
        int k = 4 * s + kof;
        float x0 = agg1[(size_t)(m0 + r16) * HID_C + k]     + b1[k];
        float x1 = agg1[(size_t)(m0 + r16) * HID_C + k + 1] + b1[k + 1];
        a[s].x = fmaxf(x0, 0.0f);
        a[s].y = fmaxf(x1, 0.0f);
    }

    const int row_off = (lane >> 4) << 3;
    #pragma unroll
    for (int t = 0; t < 3; ++t) {
        v8f acc = {};
        #pragma unroll
        for (int s = 0; s < 4; ++s) {
            int k = 4 * s + kof;
            v2f b;
            b.x = sW[k * OUT_P + t * 16 + r16];
            b.y = sW[(k + 1) * OUT_P + t * 16 + r16];
            acc = __builtin_amdgcn_wmma_f32_16x16x4_f32(false, a[s], false, b,
                                                        (short)0, acc, false, false);
        }
        int col = t * 16 + r16;
        if (col < OUT_C) {
            #pragma unroll
            for (int g = 0; g < 8; ++g) {
                int   row = m0 + g + row_off;
                float v   = acc[g];
                float di  = dinv[row];
                h2[(size_t)row * OUT_C + col]  = v;
                out[(size_t)row * OUT_C + col] = di * di * v;   // self-loop term
            }
        }
    }
}

// ---- edge aggregation, layer 2: out[col] += norm * h2[row] -----------------
__global__ __launch_bounds__(256)
void k_agg2(const long long* __restrict__ ei, const float* __restrict__ dinv,
            const float* __restrict__ h2, float* __restrict__ out, int E)
{
    const int total  = E * OUT_C;
    const int stride = gridDim.x * blockDim.x;
    for (int idx = blockIdx.x * blockDim.x + threadIdx.x; idx < total; idx += stride) {
        int e = idx / OUT_C;
        int c = idx - e * OUT_C;
        int r = (int)ei[e];
        int t = (int)ei[(size_t)E + e];
        float nrm = dinv[r] * dinv[t];
        atomAddF(&out[(size_t)t * OUT_C + c], nrm * h2[(size_t)r * OUT_C + c]);
    }
}

// ---- final: out = log_softmax(out + b2) in place ---------------------------
__global__ __launch_bounds__(256)
void k_logsoftmax(float* __restrict__ out, const float* __restrict__ b2, int n)
{
    int i = blockIdx.x * blockDim.x + threadIdx.x;
    if (i >= n) return;
    float v[OUT_C];
    float m = -3.402823466e38f;
    #pragma unroll
    for (int c = 0; c < OUT_C; ++c) {
        v[c] = out[(size_t)i * OUT_C + c] + b2[c];
        m = fmaxf(m, v[c]);
    }
    float s = 0.0f;
    #pragma unroll
    for (int c = 0; c < OUT_C; ++c) s += __expf(v[c] - m);
    float ls = __logf(s);
    #pragma unroll
    for (int c = 0; c < OUT_C; ++c) out[(size_t)i * OUT_C + c] = v[c] - m - ls;
}

// ---------------------------------------------------------------------------
extern "C" void kernel_launch(void* const* d_in, const int* in_sizes, int n_in,
                              void* d_out, int out_size, void* d_ws, size_t ws_size,
                              hipStream_t stream)
{
    const float*     x  = (const float*)d_in[0];
    const long long* ei = (const long long*)d_in[1];   // int64 [2, E]
    const float*     W1 = (const float*)d_in[2];
    const float*     b1 = (const float*)d_in[3];
    const float*     W2 = (const float*)d_in[4];
    const float*     b2 = (const float*)d_in[5];
    float*           out = (float*)d_out;

    const int N = in_sizes[0] / IN_C;
    const int E = in_sizes[1] / 2;

    // workspace layout (256B-aligned regions), ~29.3 MB total
    float* dinv = (float*)d_ws;                                   // N (deg in place)
    float* h1   = dinv + (((size_t)N + 63) & ~(size_t)63);        // N*16
    float* agg1 = h1   + (((size_t)N * HID_C + 63) & ~(size_t)63);// N*16
    float* h2   = agg1 + (((size_t)N * HID_C + 63) & ~(size_t)63);// N*40

    k_deg_init  <<<(N + 255) / 256, 256, 0, stream>>>(dinv, N);
    k_deg_count <<<(E + 255) / 256, 256, 0, stream>>>(ei, dinv, E);
    k_rsqrt     <<<(N + 255) / 256, 256, 0, stream>>>(dinv, N);
    k_gemm1     <<<(N + 127) / 128, 256, 0, stream>>>(x, W1, dinv, h1, agg1, N);
    k_agg1      <<<4096,            256, 0, stream>>>(ei, dinv, h1, agg1, E);
    k_gemm2     <<<(N + 127) / 128, 256, 0, stream>>>(agg1, b1, W2, dinv, h2, out, N);
    k_agg2      <<<8192,            256, 0, stream>>>(ei, dinv, h2, out, E);
    k_logsoftmax<<<(N + 255) / 256, 256, 0, stream>>>(out, b2, N);
}